// GAT_32195074851206
// MI455X (gfx1250) — compile-verified
//
#include <hip/hip_runtime.h>
#include <math.h>

// ---------------- problem constants (match reference) ----------------
#define N_NODES 50000
#define N_EDGES 800000
#define FIN     128
#define FHID    64
#define FOUT    16
#define NEG_SLOPE 0.2f
#define ET (N_EDGES + N_NODES)   // edges + self loops

typedef __attribute__((ext_vector_type(2))) float v2f;
typedef __attribute__((ext_vector_type(8))) float v8f;

// ---------------- utility kernels ----------------
__global__ void fill_f32(float* __restrict__ p, float v, int n) {
  int i = blockIdx.x * blockDim.x + threadIdx.x;
  if (i < n) p[i] = v;
}

__device__ __forceinline__ void atomic_max_f32(float* addr, float val) {
  // exact float max via integer ordering trick (works for mixed signs,
  // buffer initialized to -inf)
  if (val >= 0.0f)
    atomicMax((int*)addr, __float_as_int(val));
  else
    atomicMin((unsigned int*)addr, __float_as_uint(val));
}

// ---------------- fused dual GEMM with fp32 WMMA ----------------
// Yl = X @ Wl + bl ; Yr = X @ Wr + br   (X: N_NODES x K, W: K x F)
// One wave owns one 16-row block of X and ALL 2*(F/16) 16x16 output tiles,
// so the streaming A fragment is loaded once per k-step and reused for every
// column tile of both weight matrices. V_WMMA_F32_16X16X4_F32 throughout.
// N_NODES = 3125 * 16 exactly -> no partial tiles; exits are wave-uniform so
// EXEC is all-ones for every WMMA.
template <int K, int F>
__global__ __launch_bounds__(256) void gemm2_bias_wmma(
    const float* __restrict__ X,
    const float* __restrict__ Wl, const float* __restrict__ bl,
    const float* __restrict__ Wr, const float* __restrict__ br,
    float* __restrict__ Yl, float* __restrict__ Yr) {
  const int lane = threadIdx.x & 31;
  const int wave = (blockIdx.x * blockDim.x + threadIdx.x) >> 5;
  constexpr int NT = F / 16;            // col tiles per weight matrix
  constexpr int MT = N_NODES / 16;      // row tiles (exact)
  if (wave >= MT) return;               // wave-uniform exit
  const int tm  = wave;
  const int r16 = lane & 15;            // A row / B col within tile
  const int hi  = lane >> 4;            // 0 -> K pair {0,1}, 1 -> {2,3}

  const float* xrow = X + (size_t)(tm * 16 + r16) * K;

  v8f acc[2 * NT];
#pragma unroll
  for (int t = 0; t < 2 * NT; ++t) acc[t] = (v8f){};

  for (int k0 = 0; k0 < K; k0 += 4) {
    const int kk = k0 + (hi << 1);      // this half-wave's K pair
    v2f a;
    a.x = xrow[kk];
    a.y = xrow[kk + 1];
#pragma unroll
    for (int t = 0; t < NT; ++t) {
      const float* wl = Wl + (size_t)kk * F + t * 16 + r16;
      v2f b;
      b.x = wl[0];                      // W[kk][n]
      b.y = wl[F];                      // W[kk+1][n]
      acc[t] = __builtin_amdgcn_wmma_f32_16x16x4_f32(
          false, a, false, b, (short)0, acc[t], false, false);
      const float* wr = Wr + (size_t)kk * F + t * 16 + r16;
      b.x = wr[0];
      b.y = wr[F];
      acc[NT + t] = __builtin_amdgcn_wmma_f32_16x16x4_f32(
          false, a, false, b, (short)0, acc[NT + t], false, false);
    }
  }

  // C/D layout: VGPR r holds row M = r + 8*(lane>=16), col N = lane&15
  const int rowBase = tm * 16 + hi * 8;
#pragma unroll
  for (int t = 0; t < NT; ++t) {
    const float bvl = bl[t * 16 + r16];
    const float bvr = br[t * 16 + r16];
#pragma unroll
    for (int r = 0; r < 8; ++r) {
      Yl[(size_t)(rowBase + r) * F + t * 16 + r16] = acc[t][r] + bvl;
      Yr[(size_t)(rowBase + r) * F + t * 16 + r16] = acc[NT + t][r] + bvr;
    }
  }
}

// ---------------- per-edge attention logit + segment max ----------------
// One wave per edge: e = a . leaky_relu(hl[src] + hr[dst]); atomic max into m[dst]
template <int F>
__global__ __launch_bounds__(256) void edge_logit(
    const float* __restrict__ hl, const float* __restrict__ hr,
    const float* __restrict__ att, const int* __restrict__ src,
    const int* __restrict__ dst, float* __restrict__ ebuf,
    float* __restrict__ m) {
  const int lane = threadIdx.x & 31;
  const int wave = (blockIdx.x * blockDim.x + threadIdx.x) >> 5;
  if (wave >= ET) return;
  int s, d;
  if (wave < N_EDGES) { s = src[wave]; d = dst[wave]; }
  else                { s = d = wave - N_EDGES; }   // self loop

  float partial = 0.f;
#pragma unroll
  for (int f = lane; f < F; f += 32) {
    float z = hl[(size_t)s * F + f] + hr[(size_t)d * F + f];
    z = (z > 0.f) ? z : NEG_SLOPE * z;
    partial += att[f] * z;
  }
#pragma unroll
  for (int off = 16; off > 0; off >>= 1)
    partial += __shfl_xor(partial, off, 32);
  if (lane == 0) {
    ebuf[wave] = partial;
    atomic_max_f32(m + d, partial);
  }
}

// ---------------- exp + segment sum (in-place e -> exp(e - m[dst])) ----------------
__global__ __launch_bounds__(256) void edge_exp(
    float* __restrict__ ebuf, const float* __restrict__ m,
    const int* __restrict__ dst, float* __restrict__ ssum) {
  const int e = blockIdx.x * blockDim.x + threadIdx.x;
  if (e >= ET) return;
  const int d = (e < N_EDGES) ? dst[e] : (e - N_EDGES);
  const float ex = expf(ebuf[e] - m[d]);
  ebuf[e] = ex;
  atomicAdd(ssum + d, ex);
}

// ---------------- weighted scatter-aggregate ----------------
template <int F>
__global__ __launch_bounds__(256) void edge_agg(
    const float* __restrict__ hl, const float* __restrict__ exbuf,
    const float* __restrict__ ssum, const int* __restrict__ src,
    const int* __restrict__ dst, float* __restrict__ agg) {
  const int lane = threadIdx.x & 31;
  const int wave = (blockIdx.x * blockDim.x + threadIdx.x) >> 5;
  if (wave >= ET) return;
  int s, d;
  if (wave < N_EDGES) { s = src[wave]; d = dst[wave]; }
  else                { s = d = wave - N_EDGES; }
  const float alpha = exbuf[wave] / (ssum[d] + 1e-16f);
#pragma unroll
  for (int f = lane; f < F; f += 32)
    atomicAdd(&agg[(size_t)d * F + f], alpha * hl[(size_t)s * F + f]);
}

// ---------------- bias (+optional relu) finalize, in place ----------------
template <int F, bool RELU>
__global__ __launch_bounds__(256) void finalize_bias(
    float* __restrict__ y, const float* __restrict__ bias) {
  const int i = blockIdx.x * blockDim.x + threadIdx.x;
  if (i >= N_NODES * F) return;
  float v = y[i] + bias[i & (F - 1)];   // F is a power of two
  if (RELU) v = fmaxf(v, 0.f);
  y[i] = v;
}

// ---------------- host-side launcher ----------------
extern "C" void kernel_launch(void* const* d_in, const int* in_sizes, int n_in,
                              void* d_out, int out_size, void* d_ws, size_t ws_size,
                              hipStream_t stream) {
  (void)in_sizes; (void)n_in; (void)out_size; (void)ws_size;

  const float* x    = (const float*)d_in[0];
  const int*   ei   = (const int*)d_in[1];
  const int*   src  = ei;                 // edge_index[0]
  const int*   dst  = ei + N_EDGES;       // edge_index[1]
  const float* W1l  = (const float*)d_in[2];
  const float* W1r  = (const float*)d_in[3];
  const float* b1l  = (const float*)d_in[4];
  const float* b1r  = (const float*)d_in[5];
  const float* a1   = (const float*)d_in[6];
  const float* bi1  = (const float*)d_in[7];
  const float* W2l  = (const float*)d_in[8];
  const float* W2r  = (const float*)d_in[9];
  const float* b2l  = (const float*)d_in[10];
  const float* b2r  = (const float*)d_in[11];
  const float* a2   = (const float*)d_in[12];
  const float* bi2  = (const float*)d_in[13];
  float* out = (float*)d_out;

  // workspace carve-out (floats); total ~12.15M floats (~48.6 MB)
  float* ws   = (float*)d_ws;
  float* h1l  = ws;                                   // N*64
  float* h1r  = h1l + (size_t)N_NODES * FHID;         // N*64
  float* agg1 = h1r + (size_t)N_NODES * FHID;         // N*64 (becomes h after relu)
  float* h2l  = agg1 + (size_t)N_NODES * FHID;        // N*16
  float* h2r  = h2l + (size_t)N_NODES * FOUT;         // N*16
  float* ebuf = h2r + (size_t)N_NODES * FOUT;         // ET
  float* mbuf = ebuf + (size_t)ET;                    // N
  float* sbuf = mbuf + (size_t)N_NODES;               // N

  const int BLK = 256;
  const int wavesPerBlk = BLK / 32;
  const int edgeWaveGrid = (ET + wavesPerBlk - 1) / wavesPerBlk;
  const int edgeThreadGrid = (ET + BLK - 1) / BLK;
  const int gemmGrid = (N_NODES / 16 + wavesPerBlk - 1) / wavesPerBlk;  // 3125 waves
  const float NEG_INF = -__builtin_huge_valf();

  // ================= layer 1 =================
  fill_f32<<<(N_NODES + BLK - 1) / BLK, BLK, 0, stream>>>(mbuf, NEG_INF, N_NODES);
  fill_f32<<<(N_NODES + BLK - 1) / BLK, BLK, 0, stream>>>(sbuf, 0.f, N_NODES);
  fill_f32<<<(N_NODES * FHID + BLK - 1) / BLK, BLK, 0, stream>>>(agg1, 0.f, N_NODES * FHID);

  gemm2_bias_wmma<FIN, FHID><<<gemmGrid, BLK, 0, stream>>>(
      x, W1l, b1l, W1r, b1r, h1l, h1r);

  edge_logit<FHID><<<edgeWaveGrid, BLK, 0, stream>>>(h1l, h1r, a1, src, dst, ebuf, mbuf);
  edge_exp<<<edgeThreadGrid, BLK, 0, stream>>>(ebuf, mbuf, dst, sbuf);
  edge_agg<FHID><<<edgeWaveGrid, BLK, 0, stream>>>(h1l, ebuf, sbuf, src, dst, agg1);
  finalize_bias<FHID, true><<<(N_NODES * FHID + BLK - 1) / BLK, BLK, 0, stream>>>(agg1, bi1);

  // ================= layer 2 =================
  fill_f32<<<(N_NODES + BLK - 1) / BLK, BLK, 0, stream>>>(mbuf, NEG_INF, N_NODES);
  fill_f32<<<(N_NODES + BLK - 1) / BLK, BLK, 0, stream>>>(sbuf, 0.f, N_NODES);
  fill_f32<<<(N_NODES * FOUT + BLK - 1) / BLK, BLK, 0, stream>>>(out, 0.f, N_NODES * FOUT);

  gemm2_bias_wmma<FHID, FOUT><<<gemmGrid, BLK, 0, stream>>>(
      agg1, W2l, b2l, W2r, b2r, h2l, h2r);

  edge_logit<FOUT><<<edgeWaveGrid, BLK, 0, stream>>>(h2l, h2r, a2, src, dst, ebuf, mbuf);
  edge_exp<<<edgeThreadGrid, BLK, 0, stream>>>(ebuf, mbuf, dst, sbuf);
  edge_agg<FOUT><<<edgeWaveGrid, BLK, 0, stream>>>(h2l, ebuf, sbuf, src, dst, out);
  finalize_bias<FOUT, false><<<(N_NODES * FOUT + BLK - 1) / BLK, BLK, 0, stream>>>(out, bi2);
}